// PhiInv_54872502174405
// MI455X (gfx1250) — compile-verified
//
#include <hip/hip_runtime.h>

typedef float v2f __attribute__((ext_vector_type(2)));
typedef float v8f __attribute__((ext_vector_type(8)));

#define KCOMP   32
#define NSTEP   8        // KCOMP / 4 chained WMMA f32 16x16x4 steps
#define TOLR    1e-6f
#define MAXIT   400
#define WAVES   8        // 256-thread block = 8 wave32
#define LOG2E   1.4426950408889634f

// ---------------------------------------------------------------------------
// Prep kernel (1 block): softmax(w_logits), softplus(s_raw)+0.1, folded into
// the tables the solver needs:
//   ws[k]      = -s_k * log2(e)   (exp2 argument coefficient)
//   ws[32+k]   =  w_k             (B column 0)
//   ws[64+k]   =  w_k * s_k       (B column 1 -> p = -phi')
// ---------------------------------------------------------------------------
__global__ void phi_inv_prep_kernel(const float* __restrict__ wl,
                                    const float* __restrict__ sr,
                                    float* __restrict__ tab)
{
    const int k = threadIdx.x;
    if (k >= KCOMP) return;

    float mx = -3.4e38f;
    for (int i = 0; i < KCOMP; ++i) mx = fmaxf(mx, wl[i]);
    float Z = 0.f;
    for (int i = 0; i < KCOMP; ++i) Z += __expf(wl[i] - mx);

    const float wk = __expf(wl[k] - mx) / Z;
    const float x  = sr[k];
    const float sk = fmaxf(x, 0.f) + log1pf(__expf(-fabsf(x))) + 0.1f;

    tab[k]             = -sk * LOG2E;
    tab[KCOMP + k]     = wk;
    tab[2 * KCOMP + k] = wk * sk;
}

// ---------------------------------------------------------------------------
// Main solver: each wave32 Newton-solves 16 points. fp32 WMMA 16x16x4 does
// the K-reduction (f = e@w in C col 0, p = e@(w*s) in C col 1) on the matrix
// pipe while the trans pipe streams the exps (exp2 form: 1 mul + 1 v_exp each).
// ---------------------------------------------------------------------------
__global__ __launch_bounds__(256)
void phi_inv_wmma_kernel(const float* __restrict__ y,
                         const float* __restrict__ tab,
                         float* __restrict__ out, int n)
{
    __shared__ float xch[WAVES][32];   // per-wave: [0..15]=f_m, [16..31]=p_m

    const int tid  = threadIdx.x;
    const int wv   = tid >> 5;
    const int lane = tid & 31;
    const int m    = lane & 15;   // point row (A) == column index (B/C)
    const int h    = lane >> 4;   // lane half selects K-pair / C row-half

    // Per-lane tables. For WMMA step j, VGPR v: global K index = 4j + 2h + v
    // (identical formula for the gfx1250 A and B layouts).
    const float* ctab = tab;                         // -s*log2e
    const float* bsel = (m == 0) ? tab + KCOMP       // w
                                 : tab + 2 * KCOMP;  // w*s
    float cA0[NSTEP], cA1[NSTEP], B0[NSTEP], B1[NSTEP];
#pragma unroll
    for (int j = 0; j < NSTEP; ++j) {
        const int k0 = 4 * j + 2 * h;
        cA0[j] = ctab[k0];
        cA1[j] = ctab[k0 + 1];
        B0[j]  = (m < 2) ? bsel[k0]     : 0.0f;
        B1[j]  = (m < 2) ? bsel[k0 + 1] : 0.0f;
    }

    const long wave_global = (long)blockIdx.x * (blockDim.x >> 5) + wv;
    const long pidx   = wave_global * 16 + m;
    const long pclamp = (pidx < n) ? pidx : (long)(n - 1);
    const float yv = y[pclamp];

    float t = 0.0f, f = 0.0f, p = 1.0f;

    // phi and -phi' at t, via 8 chained fp32 WMMAs + LDS column exchange.
    auto evalFP = [&](float tt, float& fo, float& po) {
        v8f C = {};
#pragma unroll
        for (int j = 0; j < NSTEP; ++j) {
            v2f A, Bv;
            A.x  = __builtin_amdgcn_exp2f(tt * cA0[j]);  // exp(-t*s_k)
            A.y  = __builtin_amdgcn_exp2f(tt * cA1[j]);
            Bv.x = B0[j];
            Bv.y = B1[j];
            // D = A(16x4) * B(4x16) + C ; exact fp32 accumulation
            C = __builtin_amdgcn_wmma_f32_16x16x4_f32(
                    false, A, false, Bv, (short)0, C, false, false);
        }
        // C col 0 (f) lives in lanes 0/16, col 1 (p) in lanes 1/17;
        // VGPR r holds rows M=r (lanes 0-15) and M=r+8 (lanes 16-31).
        if (m < 2) {
            float* dst = &xch[wv][m * 16 + h * 8];
#pragma unroll
            for (int r = 0; r < 8; ++r) dst[r] = C[r];
        }
        __builtin_amdgcn_wave_barrier();   // keep DS stores before DS loads
        fo = xch[wv][m];
        po = xch[wv][16 + m];
        __builtin_amdgcn_wave_barrier();   // keep loads before next iter stores
    };

    // Newton loop: wave-uniform exit (ballot) so EXEC is all-ones at every WMMA.
    for (int it = 0; it < MAXIT; ++it) {
        evalFP(t, f, p);
        const float g = f - yv;
        t = __builtin_fmaf(g, __builtin_amdgcn_rcpf(p), t);  // t - g/phi'
        const unsigned notdone =
            __builtin_amdgcn_ballot_w32(fabsf(g) >= TOLR);
        if (notdone == 0u) break;
    }

    // Implicit-function correction == one more Newton step at t*.
    evalFP(t, f, p);
    const float res = __builtin_fmaf(f - yv, __builtin_amdgcn_rcpf(p), t);

    if (h == 0 && pidx < (long)n) out[pidx] = res;       // lanes 0-15 own points
}

extern "C" void kernel_launch(void* const* d_in, const int* in_sizes, int n_in,
                              void* d_out, int out_size, void* d_ws, size_t ws_size,
                              hipStream_t stream) {
    const float* y  = (const float*)d_in[0];
    const float* wl = (const float*)d_in[1];
    const float* sr = (const float*)d_in[2];
    float* out = (float*)d_out;
    float* tab = (float*)d_ws;          // 96 floats used
    const int n = in_sizes[0];

    phi_inv_prep_kernel<<<1, 32, 0, stream>>>(wl, sr, tab);

    // 16 points per wave32, 8 waves per block -> 128 points per block
    const int pointsPerBlock = 16 * WAVES;
    const int grid = (n + pointsPerBlock - 1) / pointsPerBlock;
    phi_inv_wmma_kernel<<<grid, 256, 0, stream>>>(y, tab, out, n);
}